// TrainingLoss_28767690949498
// MI455X (gfx1250) — compile-verified
//
#include <hip/hip_runtime.h>
#include <hip/hip_bf16.h>

// ---------------- problem constants ----------------
#define HH   320
#define WWD  480
#define HWF  153600      // 320*480
#define KC   26
#define CH   256
#define SH   80
#define SW   120
#define SHW  9600        // 80*120
#define NM   234         // 9*26 memory entries
#define NMP  240         // padded to 15 col-tiles of 16
#define NPIX_S 19200     // 2*9600

// ---------------- workspace layout (float offsets) ----------------
#define WS_CE   0
#define WS_MET  1
#define WS_CNT  2                       // 26 floats (class counts)
#define WS_PSUM 28                      // 3 * 26*256 proto sums
#define PSUM_SZ (KC*CH)                 // 6656
#define WS_INVA (WS_PSUM + 3*PSUM_SZ)   // 19996 ; 3*2*9600 inv row norms
#define WS_CMAT (WS_INVA + 3*2*SHW)     // 77596 ; 2*256*240 normalized C
#define WS_TOT  (WS_CMAT + 2*CH*NMP)    // 200476 floats (~0.8 MB)

typedef float v2f __attribute__((ext_vector_type(2)));
typedef float v8f __attribute__((ext_vector_type(8)));

// ---------------- zero the accumulator region ----------------
__global__ void k_zero(float* ws, int n) {
    int i = blockIdx.x * blockDim.x + threadIdx.x;
    if (i < n) ws[i] = 0.f;
}

// ---------------- CE: sum of -log_softmax picks over 4 tensors ----------------
__global__ void k_ce(const float* p0, const float* p1, const float* p2, const float* p3,
                     const int* labels, float* acc) {
    const float* ps[4] = {p0, p1, p2, p3};
    const float* pr = ps[blockIdx.y];
    int pix = blockIdx.x * blockDim.x + threadIdx.x;   // 0..307199 (exact grid)
    int b = pix / HWF;
    int s = pix % HWF;
    int lab = labels[(size_t)b * 3 * HWF + s];         // labels[:,0,0]
    const float* base = pr + ((size_t)(b * 3 + 2)) * KC * HWF + s;  // t = 2 slice
    float vals[KC];
    float mx = -3.0e38f, picked = 0.f;
#pragma unroll
    for (int k = 0; k < KC; k++) {
        float v = base[(size_t)k * HWF];
        vals[k] = v;
        mx = fmaxf(mx, v);
        picked = (k == lab) ? v : picked;
    }
    float se = 0.f;
#pragma unroll
    for (int k = 0; k < KC; k++) se += expf(vals[k] - mx);
    float nll = -(picked - mx - logf(se));
    // block reduce -> 1 atomic per block
    __shared__ float red[8];
    float v = nll;
    for (int off = 16; off >= 1; off >>= 1) v += __shfl_down(v, off, 32);
    int lane = threadIdx.x & 31, wv = threadIdx.x >> 5;
    if (lane == 0) red[wv] = v;
    __syncthreads();
    if (wv == 0) {
        float t = (lane < 8) ? red[lane] : 0.f;
        for (int off = 4; off >= 1; off >>= 1) t += __shfl_down(t, off, 32);
        if (lane == 0) atomicAdd(acc, t);
    }
}

// ---------------- prototype scatter-sums (pooled over both batches) ----------------
__global__ void k_proto(const float* fR, const float* fT, const float* fF,
                        const int* labels, float* ws) {
    int x = blockIdx.y;
    const float* ft = (x == 0) ? fR : ((x == 1) ? fT : fF);
    __shared__ float lsum[KC * CH];
    __shared__ float lcnt[KC];
    int cf = threadIdx.x;                      // 256 threads = one channel each
#pragma unroll
    for (int k = 0; k < KC; k++) lsum[k * CH + cf] = 0.f;
    if (cf < KC) lcnt[cf] = 0.f;
    __syncthreads();
    int p0 = blockIdx.x * 192;                 // 100 blocks * 192 px = 19200 exact
    for (int q = 0; q < 192; q++) {
        int p = p0 + q;
        int b = p / SHW, pp = p % SHW;
        int i = pp / SW, j = pp % SW;
        int lab = labels[(size_t)b * 3 * HWF + (4 * i) * WWD + 4 * j]; // y_small
        float v = ft[((size_t)(b * CH + cf)) * SHW + pp];
        lsum[lab * CH + cf] += v;              // unique channel per thread: race-free
        if (cf == 0) lcnt[lab] += 1.f;
    }
    __syncthreads();
    float* psum = ws + WS_PSUM + x * PSUM_SZ;
#pragma unroll
    for (int k = 0; k < KC; k++) atomicAdd(&psum[k * CH + cf], lsum[k * CH + cf]);
    if (x == 0 && cf < KC) atomicAdd(&ws[WS_CNT + cf], lcnt[cf]);
}

// ---------------- anchor row 1/||a|| (coalesced: thread = pixel) ----------------
__global__ void k_invnorm(const float* fR, const float* fT, const float* fF, float* ws) {
    int id = blockIdx.x * blockDim.x + threadIdx.x;   // < 57600 exact
    int x = id / NPIX_S;
    int rem = id % NPIX_S;
    int b = rem / SHW, p = rem % SHW;
    const float* base = ((x == 0) ? fR : ((x == 1) ? fT : fF)) + (size_t)b * CH * SHW + p;
    float ss = 0.f;
    for (int c = 0; c < CH; c++) { float v = base[(size_t)c * SHW]; ss += v * v; }
    ws[WS_INVA + id] = 1.0f / fmaxf(sqrtf(ss), 1e-12f);
}

// ---------------- build normalized C, cf-major (WMMA B-friendly) ----------------
__global__ void k_cbuild(const float* mR, const float* mT, const float* mF, float* ws) {
    int wv = (blockIdx.x * blockDim.x + threadIdx.x) >> 5;  // 0..479
    int lane = threadIdx.x & 31;
    int b = wv / NMP, m = wv % NMP;
    float* cmat = ws + WS_CMAT + (size_t)b * CH * NMP;
    if (m >= NM) {  // zero pad columns (uniform per wave)
        for (int i = 0; i < 8; i++) cmat[(size_t)(lane + 32 * i) * NMP + m] = 0.f;
        return;
    }
    int t3 = m / KC, k = m % KC;   // reshape order: m = t3*K + k
    float vals[8];
    if (t3 == 2 || t3 == 5 || t3 == 8) {           // prototype slots (batch-shared)
        int x = t3 / 3;
        float cnt = fmaxf(ws[WS_CNT + k], 1.0f);
        const float* psum = ws + WS_PSUM + x * PSUM_SZ;
        for (int i = 0; i < 8; i++) vals[i] = psum[k * CH + (lane + 32 * i)] / cnt;
    } else {                                        // memory bank slots
        const float* mm = (t3 < 3) ? mR : ((t3 < 6) ? mT : mF);
        int tm = t3 % 3;  // 0 or 1
        for (int i = 0; i < 8; i++) {
            int cf = lane + 32 * i;
            vals[i] = mm[(((size_t)(b * CH + cf)) * 2 + tm) * KC + k];
        }
    }
    float ss = 0.f;
    for (int i = 0; i < 8; i++) ss += vals[i] * vals[i];
    for (int off = 16; off >= 1; off >>= 1) ss += __shfl_xor(ss, off, 32);
    float rn = 1.0f / fmaxf(sqrtf(ss), 1e-12f);
    for (int i = 0; i < 8; i++) cmat[(size_t)(lane + 32 * i) * NMP + m] = vals[i] * rn;
}

// ---------------- fused WMMA GEMM + contrastive epilogue ----------------
// wave = one 16-row tile of one (anchor, batch); 15 col-tiles x 64 k-steps of
// v_wmma_f32_16x16x4_f32; epilogue uses the 16x16 f32 D layout:
// VGPR r, lanes 0-15 -> (M=r, N=lane); lanes 16-31 -> (M=8+r, N=lane-16).
__global__ void __launch_bounds__(128) k_metric(
        const float* fR, const float* fT, const float* fF,
        const int* labels, const float* inva_ws, const float* cmat_ws,
        float* metric_acc) {
    int wv = (blockIdx.x * blockDim.x + threadIdx.x) >> 5;  // 0..3599 exact
    int lane = threadIdx.x & 31;
    int x = wv / 1200;
    int rem = wv % 1200;
    int b = rem / 600, tile = rem % 600;
    int row0 = tile * 16;
    const float* A = ((x == 0) ? fR : ((x == 1) ? fT : fF)) + (size_t)b * CH * SHW;
    const float* Cm = cmat_ws + (size_t)b * CH * NMP;
    const float* inva = inva_ws + ((size_t)x * 2 + b) * SHW;
    int half = lane >> 4, l15 = lane & 15;

    v8f acc[15];
#pragma unroll
    for (int n = 0; n < 15; n++)
#pragma unroll
        for (int r = 0; r < 8; r++) acc[n][r] = 0.f;

    // K loop: A frag = 16x4 f32 (lanes 0-15: K=cf0,cf0+1 ; 16-31: cf0+2,cf0+3)
    for (int kk = 0; kk < 64; kk++) {
        int cf = kk * 4 + 2 * half;
        v2f va;
        va.x = A[(size_t)cf * SHW + row0 + l15];          // coalesced per 16-lane half
        va.y = A[(size_t)(cf + 1) * SHW + row0 + l15];
#pragma unroll
        for (int n = 0; n < 15; n++) {
            v2f vb;
            vb.x = Cm[(size_t)cf * NMP + n * 16 + l15];
            vb.y = Cm[(size_t)(cf + 1) * NMP + n * 16 + l15];
            acc[n] = __builtin_amdgcn_wmma_f32_16x16x4_f32(
                false, va, false, vb, (short)0, acc[n], false, false);
        }
    }

    // epilogue: per VGPR slot r, lanes 0-15 own row row0+r, lanes 16-31 row0+8+r
    float wave_sum = 0.f;
#pragma unroll
    for (int r = 0; r < 8; r++) {
        int row = row0 + 8 * half + r;
        float s = inva[row] * 10.0f;   // 1/||a|| * 1/TEMP
        int lab = labels[(size_t)b * 3 * HWF + (4 * (row / SW)) * WWD + 4 * (row % SW)];
        float t[15];
        float mx = -3.0e38f;
#pragma unroll
        for (int n = 0; n < 15; n++) {
            int m = n * 16 + l15;
            float v = (m < NM) ? acc[n][r] * s : -3.0e38f;
            t[n] = v;
            mx = fmaxf(mx, v);
        }
        for (int off = 8; off >= 1; off >>= 1) mx = fmaxf(mx, __shfl_xor(mx, off, 32));
        int mlo = 9 * lab, mhi = mlo + 9;   // mask class = m // 9 (as reference)
        float neg = 0.f;
#pragma unroll
        for (int n = 0; n < 15; n++) {
            int m = n * 16 + l15;
            if (m < NM && !(m >= mlo && m < mhi)) neg += expf(t[n] - mx);
        }
        for (int off = 8; off >= 1; off >>= 1) neg += __shfl_xor(neg, off, 32);
        float pos = 0.f;
#pragma unroll
        for (int n = 0; n < 15; n++) {
            int m = n * 16 + l15;
            if (m >= mlo && m < mhi) {
                float ts = t[n] - mx;
                pos += ts - logf(expf(ts) + neg);   // log_prob of positive entry
            }
        }
        for (int off = 8; off >= 1; off >>= 1) pos += __shfl_xor(pos, off, 32);
        if (l15 == 0) wave_sum += pos;
    }
    if (l15 == 0) atomicAdd(metric_acc, wave_sum);
}

// ---------------- finalize scalar ----------------
__global__ void k_final(const float* ws, float* out) {
    float ce = ws[WS_CE] / (4.0f * 2.0f * (float)HWF);          // 4 tensors * B*H*W
    float met = -ws[WS_MET] / (9.0f * (float)SHW * 2.0f * 3.0f); // /mask9 /p /b /anchors
    out[0] = ce + 0.001f * met;
}

extern "C" void kernel_launch(void* const* d_in, const int* in_sizes, int n_in,
                              void* d_out, int out_size, void* d_ws, size_t ws_size,
                              hipStream_t stream) {
    const float* probs  = (const float*)d_in[0];
    const int*   labels = (const int*)d_in[1];
    const float* paux   = (const float*)d_in[2];
    const float* pth    = (const float*)d_in[3];
    const float* pfu    = (const float*)d_in[4];
    const float* fR     = (const float*)d_in[5];
    const float* fT     = (const float*)d_in[6];
    const float* fF     = (const float*)d_in[7];
    const float* mR     = (const float*)d_in[8];
    const float* mT     = (const float*)d_in[9];
    const float* mF     = (const float*)d_in[10];
    float* ws  = (float*)d_ws;
    float* out = (float*)d_out;

    k_zero<<<dim3((WS_INVA + 255) / 256), 256, 0, stream>>>(ws, WS_INVA);
    k_ce<<<dim3(1200, 4), 256, 0, stream>>>(probs, paux, pth, pfu, labels, ws + WS_CE);
    k_proto<<<dim3(100, 3), 256, 0, stream>>>(fR, fT, fF, labels, ws);
    k_invnorm<<<225, 256, 0, stream>>>(fR, fT, fF, ws);
    k_cbuild<<<120, 128, 0, stream>>>(mR, mT, mF, ws);
    k_metric<<<900, 128, 0, stream>>>(fR, fT, fF, labels, ws + WS_INVA, ws + WS_CMAT,
                                      ws + WS_MET);
    k_final<<<1, 1, 0, stream>>>(ws, out);
}